// MambaMFNet_1202590843335
// MI455X (gfx1250) — compile-verified
//
#include <hip/hip_runtime.h>
#include <math.h>

// ---------------- model dims ----------------
#define DMODEL 128
#define RD     64
#define DINNER 128
#define DSTATE 16
#define DTRANK 4
#define BSZ    2
#define HH     128
#define WW     128
#define LSEQ   (HH * WW)          // 16384
#define BL     (BSZ * LSEQ)       // 32768
#define EPSLN  1e-5f

typedef __attribute__((ext_vector_type(2))) float v2f;
typedef __attribute__((ext_vector_type(8))) float v8f;

__device__ __forceinline__ float wave_sum(float v) {
  #pragma unroll
  for (int o = 16; o >= 1; o >>= 1) v += __shfl_xor(v, o, 32);
  return v;
}

// ============ K1: LayerNorm over 128 channels, one wave per row ============
__global__ void ln128_kernel(const float* __restrict__ x, const float* __restrict__ w,
                             const float* __restrict__ b, float* __restrict__ y) {
  int wave = threadIdx.x >> 5, lane = threadIdx.x & 31;
  long row = (long)blockIdx.x * 8 + wave;
  const float4* xr = (const float4*)(x + row * DMODEL);
  float4 v = xr[lane];
  float s = v.x + v.y + v.z + v.w;
  float q = v.x*v.x + v.y*v.y + v.z*v.z + v.w*v.w;
  s = wave_sum(s); q = wave_sum(q);
  float mu  = s * (1.0f / DMODEL);
  float var = q * (1.0f / DMODEL) - mu * mu;
  float rs  = rsqrtf(var + EPSLN);
  float4 wv = ((const float4*)w)[lane];
  float4 bv = ((const float4*)b)[lane];
  float4 o;
  o.x = (v.x - mu) * rs * wv.x + bv.x;
  o.y = (v.y - mu) * rs * wv.y + bv.y;
  o.z = (v.z - mu) * rs * wv.z + bv.z;
  o.w = (v.w - mu) * rs * wv.w + bv.w;
  ((float4*)(y + row * DMODEL))[lane] = o;
}

// ============ WMMA fp32 GEMM, fast path: N % 32 == 0 ============
// out[M,N] = act(X[M,KT] (*X2) @ W[N,KT]^T + bias) (+res)
// One wave computes a 16x32 tile: A loaded once, feeds two V_WMMA_F32_16X16X4_F32.
// All loads unconditional (no exec-mask dances); K fully unrolled with imm offsets.
template<int KT, int ACT, bool MUL, bool RES>
__global__ void gemm2_wmma_nt(const float* __restrict__ X, const float* __restrict__ X2,
                              const float* __restrict__ W, const float* __restrict__ bias,
                              const float* __restrict__ res, float* __restrict__ out,
                              int M, int N) {
  int lane = threadIdx.x & 31;
  int wave = threadIdx.x >> 5;
  int row0 = (blockIdx.x * 4 + wave) * 16;
  int col0 = blockIdx.y * 32;
  int half = lane >> 4, lo = lane & 15;

  const float* xrow  = X + (size_t)(row0 + lo) * KT + 2 * half;
  const float* x2row = MUL ? (X2 + (size_t)(row0 + lo) * KT + 2 * half) : nullptr;
  const float* wrow0 = W + (size_t)(col0 + lo) * KT + 2 * half;
  const float* wrow1 = W + (size_t)(col0 + 16 + lo) * KT + 2 * half;

  v8f acc0 = {}, acc1 = {};
  #pragma unroll
  for (int k = 0; k < KT; k += 4) {
    v2f a  = *(const v2f*)(xrow  + k);
    if (MUL) { v2f m = *(const v2f*)(x2row + k); a.x *= m.x; a.y *= m.y; }
    v2f b0 = *(const v2f*)(wrow0 + k);
    v2f b1 = *(const v2f*)(wrow1 + k);
    acc0 = __builtin_amdgcn_wmma_f32_16x16x4_f32(false, a, false, b0, (short)0, acc0, false, false);
    acc1 = __builtin_amdgcn_wmma_f32_16x16x4_f32(false, a, false, b1, (short)0, acc1, false, false);
  }

  float bv0 = bias ? bias[col0 + lo]      : 0.0f;
  float bv1 = bias ? bias[col0 + 16 + lo] : 0.0f;
  #pragma unroll
  for (int r = 0; r < 8; r++) {
    int row = row0 + r + 8 * half;                      // D layout: VGPR r -> M=r / M=r+8
    float v0 = acc0[r] + bv0;
    float v1 = acc1[r] + bv1;
    if (ACT == 1) {
      v0 = 0.5f * v0 * (1.0f + erff(v0 * 0.70710678118654752f));
      v1 = 0.5f * v1 * (1.0f + erff(v1 * 0.70710678118654752f));
    }
    size_t o0 = (size_t)row * N + col0 + lo;
    if (RES) { v0 += res[o0]; v1 += res[o0 + 16]; }
    out[o0]      = v0;
    out[o0 + 16] = v1;
  }
}

// ============ WMMA fp32 GEMM, masked path (ragged N; used for N=36) ============
// Unconditional B loads through a clamped row pointer, zeroed via 0/1 mask multiply.
template<int KT>
__global__ void gemm_wmma_nt_masked(const float* __restrict__ X, const float* __restrict__ W,
                                    float* __restrict__ out, int M, int N) {
  int lane = threadIdx.x & 31;
  int wave = threadIdx.x >> 5;
  int row0 = (blockIdx.x * 4 + wave) * 16;
  int col0 = blockIdx.y * 16;
  int half = lane >> 4, lo = lane & 15;

  int  bcol  = col0 + lo;
  float bmsk = (bcol < N) ? 1.0f : 0.0f;
  int  ccol  = (bcol < N) ? bcol : (N - 1);             // safe row, result masked to 0

  const float* xrow = X + (size_t)(row0 + lo) * KT + 2 * half;
  const float* wrow = W + (size_t)ccol * KT + 2 * half;

  v8f acc = {};
  #pragma unroll
  for (int k = 0; k < KT; k += 4) {
    v2f a = *(const v2f*)(xrow + k);
    v2f b = *(const v2f*)(wrow + k);
    b.x *= bmsk; b.y *= bmsk;
    acc = __builtin_amdgcn_wmma_f32_16x16x4_f32(false, a, false, b, (short)0, acc, false, false);
  }

  #pragma unroll
  for (int r = 0; r < 8; r++) {
    int row = row0 + r + 8 * half;
    if (bcol < N) out[(size_t)row * N + bcol] = acc[r]; // stores after all WMMA: safe
  }
}

// ============ K3: depthwise 3x3 conv (pad 1) + LayerNorm(64), one wave per (b,l) ============
__global__ void conv3x3_ln64_kernel(const float* __restrict__ p1, const float* __restrict__ cw,
                                    const float* __restrict__ cb, const float* __restrict__ lw,
                                    const float* __restrict__ lb, float* __restrict__ xs) {
  int wave = threadIdx.x >> 5, lane = threadIdx.x & 31;
  long row = (long)blockIdx.x * 8 + wave;               // row in [0, B*L)
  int b = (int)(row >> 14), l = (int)(row & (LSEQ - 1));
  int i = l >> 7, j = l & 127;
  int c0 = lane, c1 = lane + 32;
  float acc0 = cb[c0], acc1 = cb[c1];
  #pragma unroll
  for (int di = -1; di <= 1; di++) {
    int ii = i + di; if (ii < 0 || ii >= HH) continue;  // wave-uniform
    #pragma unroll
    for (int dj = -1; dj <= 1; dj++) {
      int jj = j + dj; if (jj < 0 || jj >= WW) continue;
      const float* pr = p1 + ((long)b * LSEQ + ii * WW + jj) * RD;
      int wi = (di + 1) * 3 + (dj + 1);
      acc0 += pr[c0] * cw[c0 * 9 + wi];
      acc1 += pr[c1] * cw[c1 * 9 + wi];
    }
  }
  float s = wave_sum(acc0 + acc1);
  float q = wave_sum(acc0 * acc0 + acc1 * acc1);
  float mu  = s * (1.0f / RD);
  float var = q * (1.0f / RD) - mu * mu;
  float rs  = rsqrtf(var + EPSLN);
  float* xr = xs + row * RD;
  xr[c0] = (acc0 - mu) * rs * lw[c0] + lb[c0];
  xr[c1] = (acc1 - mu) * rs * lw[c1] + lb[c1];
}

// ============ K5: causal depthwise conv1d (width 4) + SiLU ============
__global__ void conv1d_silu_kernel(const float* __restrict__ xz, const float* __restrict__ cw,
                                   const float* __restrict__ cb, float* __restrict__ xm) {
  long idx = (long)blockIdx.x * blockDim.x + threadIdx.x;   // B*L*DINNER threads
  int d = (int)(idx & (DINNER - 1));
  long row = idx >> 7;
  int b = (int)(row >> 14), l = (int)(row & (LSEQ - 1));
  float acc = cb[d];
  #pragma unroll
  for (int k = 0; k < 4; k++) {
    int ls = l - 3 + k;
    int lc = ls >= 0 ? ls : 0;                              // clamp + mask: no divergent load
    float msk = ls >= 0 ? 1.0f : 0.0f;
    acc += msk * xz[((long)b * LSEQ + lc) * (2 * DINNER) + d] * cw[d * 4 + k];
  }
  xm[row * DINNER + d] = acc / (1.0f + expf(-acc));         // SiLU
}

// ============ K7: delta = softplus(dt @ dt_proj^T + b), K=4 -> VALU ============
__global__ void delta_kernel(const float* __restrict__ dbl, const float* __restrict__ dtw,
                             const float* __restrict__ dtb, float* __restrict__ delta) {
  long idx = (long)blockIdx.x * blockDim.x + threadIdx.x;
  int d = (int)(idx & (DINNER - 1));
  long row = idx >> 7;
  const float* dr = dbl + row * (DTRANK + 2 * DSTATE);
  float t = dtb[d];
  t += dr[0] * dtw[d * 4 + 0] + dr[1] * dtw[d * 4 + 1]
     + dr[2] * dtw[d * 4 + 2] + dr[3] * dtw[d * 4 + 3];
  delta[row * DINNER + d] = (t > 20.0f) ? t : log1pf(expf(t));
}

// ============ K8: selective scan fused with y*silu(z) ============
// lane = (n, d%2) within wave; wave covers 2 d-channels; 16 blocks x 256 thr = 4096 lanes.
__global__ void selective_scan_kernel(const float* __restrict__ delta, const float* __restrict__ xm,
                                      const float* __restrict__ dbl, const float* __restrict__ xz,
                                      const float* __restrict__ A_log, const float* __restrict__ Dsk,
                                      float* __restrict__ ym) {
  int wave = threadIdx.x >> 5, lane = threadIdx.x & 31;
  int sub = lane >> 4, n = lane & 15;
  int b  = blockIdx.x >> 3;
  int d  = (blockIdx.x & 7) * 16 + wave * 2 + sub;
  float A  = -expf(A_log[d * DSTATE + n]);
  float Dv = Dsk[d];
  long base = (long)b * LSEQ;
  float h = 0.0f;
  for (int l = 0; l < LSEQ; l++) {
    long r = base + l;
    if (l + 16 < LSEQ) {                                   // hide latency of the serial chain
      __builtin_prefetch(&delta[(r + 16) * DINNER + d], 0, 0);
      __builtin_prefetch(&dbl[(r + 16) * (DTRANK + 2 * DSTATE) + DTRANK + n], 0, 0);
    }
    float dv = delta[r * DINNER + d];
    float uv = xm[r * DINNER + d];
    float Bv = dbl[r * (DTRANK + 2 * DSTATE) + DTRANK + n];
    float Cv = dbl[r * (DTRANK + 2 * DSTATE) + DTRANK + DSTATE + n];
    float dA = expf(dv * A);
    h = fmaf(dA, h, dv * Bv * uv);
    float p = h * Cv;
    p += __shfl_xor(p, 1, 32); p += __shfl_xor(p, 2, 32);
    p += __shfl_xor(p, 4, 32); p += __shfl_xor(p, 8, 32);
    if (n == 0) {
      float zv = xz[r * (2 * DINNER) + DINNER + d];
      float yv = p + uv * Dv;
      ym[r * DINNER + d] = yv * (zv / (1.0f + expf(-zv)));
    }
  }
}

// ================= launcher =================
extern "C" void kernel_launch(void* const* d_in, const int* in_sizes, int n_in,
                              void* d_out, int out_size, void* d_ws, size_t ws_size,
                              hipStream_t stream) {
  const float* x         = (const float*)d_in[0];
  const float* ln_w      = (const float*)d_in[1];
  const float* ln_b      = (const float*)d_in[2];
  const float* W1        = (const float*)d_in[3];
  const float* b1        = (const float*)d_in[4];
  const float* conv_w    = (const float*)d_in[5];
  const float* conv_b    = (const float*)d_in[6];
  const float* ssm_ln_w  = (const float*)d_in[7];
  const float* ssm_ln_b  = (const float*)d_in[8];
  const float* in_proj_w = (const float*)d_in[9];
  const float* conv1d_w  = (const float*)d_in[10];
  const float* conv1d_b  = (const float*)d_in[11];
  const float* x_proj_w  = (const float*)d_in[12];
  const float* dt_proj_w = (const float*)d_in[13];
  const float* dt_proj_b = (const float*)d_in[14];
  const float* A_log     = (const float*)d_in[15];
  const float* D_skip    = (const float*)d_in[16];
  const float* out_proj_w= (const float*)d_in[17];
  const float* W2        = (const float*)d_in[18];
  const float* b2        = (const float*)d_in[19];
  const float* W3        = (const float*)d_in[20];
  const float* b3        = (const float*)d_in[21];
  float* out = (float*)d_out;

  float* ws = (float*)d_ws;
  float* xn    = ws;                         // BL*128
  float* p1    = xn    + (size_t)BL * 128;   // BL*64
  float* p2    = p1    + (size_t)BL * 64;    // BL*64
  float* xs    = p2    + (size_t)BL * 64;    // BL*64
  float* xzb   = xs    + (size_t)BL * 64;    // BL*256
  float* xmb   = xzb   + (size_t)BL * 256;   // BL*128
  float* dblb  = xmb   + (size_t)BL * 128;   // BL*36
  float* dlt   = dblb  + (size_t)BL * 36;    // BL*128
  float* ymb   = dlt   + (size_t)BL * 128;   // BL*128
  float* p1s   = ymb   + (size_t)BL * 128;   // BL*64

  // K1: xn = LN(x)
  ln128_kernel<<<BL / 8, 256, 0, stream>>>(x, ln_w, ln_b, xn);

  // K2: p1 = xn@W1^T + b1 ; p2 = gelu(xn@W2^T + b2)
  {
    dim3 grid(BL / 64, 64 / 32);
    gemm2_wmma_nt<128, 0, false, false><<<grid, 128, 0, stream>>>(xn, nullptr, W1, b1, nullptr, p1, BL, 64);
    gemm2_wmma_nt<128, 1, false, false><<<grid, 128, 0, stream>>>(xn, nullptr, W2, b2, nullptr, p2, BL, 64);
  }

  // K3: xs = LN64(depthwise 3x3 conv(p1) + conv_b)
  conv3x3_ln64_kernel<<<BL / 8, 256, 0, stream>>>(p1, conv_w, conv_b, ssm_ln_w, ssm_ln_b, xs);

  // K4: xz = xs @ in_proj^T   [BL,256]
  {
    dim3 grid(BL / 64, 256 / 32);
    gemm2_wmma_nt<64, 0, false, false><<<grid, 128, 0, stream>>>(xs, nullptr, in_proj_w, nullptr, nullptr, xzb, BL, 256);
  }

  // K5: xm = silu(causal depthwise conv1d(xz[:, :128]))
  conv1d_silu_kernel<<<(BL * DINNER) / 256, 256, 0, stream>>>(xzb, conv1d_w, conv1d_b, xmb);

  // K6: dbl = xm @ x_proj^T   [BL,36]  (ragged N -> masked path)
  {
    dim3 grid(BL / 64, (36 + 15) / 16);
    gemm_wmma_nt_masked<128><<<grid, 128, 0, stream>>>(xmb, x_proj_w, dblb, BL, 36);
  }

  // K7: delta = softplus(dt @ dt_proj^T + dt_proj_b)
  delta_kernel<<<(BL * DINNER) / 256, 256, 0, stream>>>(dblb, dt_proj_w, dt_proj_b, dlt);

  // K8: selective scan + y*silu(z)
  selective_scan_kernel<<<BSZ * 8, 256, 0, stream>>>(dlt, xmb, dblb, xzb, A_log, D_skip, ymb);

  // K9: p1s = ym @ out_proj^T  [BL,64]
  {
    dim3 grid(BL / 64, 64 / 32);
    gemm2_wmma_nt<128, 0, false, false><<<grid, 128, 0, stream>>>(ymb, nullptr, out_proj_w, nullptr, nullptr, p1s, BL, 64);
  }

  // K10: out = (p1s * p2) @ W3^T + b3 + x
  {
    dim3 grid(BL / 64, 128 / 32);
    gemm2_wmma_nt<64, 0, true, true><<<grid, 128, 0, stream>>>(p1s, p2, W3, b3, x, out, BL, 128);
  }
}